// TopK_70007966925570
// MI455X (gfx1250) — compile-verified
//
#include <hip/hip_runtime.h>
#include <stdint.h>

// Top-k (k=64) row mask for 8192 x 24576 f32.
// One workgroup per row; row staged to LDS via async global->LDS B128 DMA
// (ASYNCcnt / s_wait_asynccnt), 4x8-bit MSD radix select on order-preserving
// uint keys (per-wave privatized LDS histograms), exact jax.lax.top_k tie
// semantics, and non-temporal coalesced 16B writeback.
// HBM traffic = 1 read + 1 write per element -> ~69 us floor at 23.3 TB/s.

#define TK_N      24576     // row length (compile-time; matches reference)
#define TK_K      64        // k (reference passes 64)
#define TK_T      1024      // threads per block (32 wave32)
#define TK_WAVES  (TK_T / 32)
#define TK_V4     (TK_N / 4 / TK_T)   // 16B chunks per thread = 6

typedef float __attribute__((ext_vector_type(4))) v4f;   // native vector for NT store

__device__ __forceinline__ uint32_t f2key(uint32_t u) {
  // monotone map: ascending uint order == ascending float order
  return (u & 0x80000000u) ? ~u : (u | 0x80000000u);
}
__device__ __forceinline__ float key2f(uint32_t k) {
  uint32_t u = (k & 0x80000000u) ? (k & 0x7fffffffu) : ~k;
  return __uint_as_float(u);
}

__global__ __launch_bounds__(TK_T, 1)
void topk_mask_kernel(const float* __restrict__ x, float* __restrict__ out) {
  __shared__ __align__(16) uint32_t keys[TK_N];          // 96 KB: whole row
  __shared__ uint32_t whist[TK_WAVES * 256];             // 32 KB: per-wave hists
  __shared__ uint32_t hist[256];
  __shared__ uint32_t suf[256];
  __shared__ uint32_t s_bin, s_above, s_bincnt;

  const int tid = threadIdx.x;
  const uint64_t rowBase = (uint64_t)blockIdx.x * TK_N;
  const float* __restrict__ xrow = x + rowBase;

  // ---- Stage row into LDS: async global->LDS, 16B per lane per issue ----
  #pragma unroll
  for (int c = 0; c < TK_V4; ++c) {
    const int v4 = tid + c * TK_T;                       // coalesced 16B chunks
    const float* g = xrow + (size_t)v4 * 4;
    uint32_t lds = (uint32_t)(uintptr_t)(&keys[v4 * 4]); // LDS byte offset
    asm volatile("global_load_async_to_lds_b128 %0, %1, off"
                 :: "v"(lds), "v"(g) : "memory");
  }
#if __has_builtin(__builtin_amdgcn_s_wait_asynccnt)
  __builtin_amdgcn_s_wait_asynccnt(0);
#else
  asm volatile("s_wait_asynccnt 0x0" ::: "memory");
#endif
  __syncthreads();   // every wave waited on its own DMAs -> row fully resident

  // ---- Convert to monotone keys in place ----
  for (int i = tid; i < TK_N; i += TK_T) keys[i] = f2key(keys[i]);
  __syncthreads();

  // ---- 4x8-bit MSD radix select of the K-th largest key ----
  uint32_t prefix = 0;       // resolved high bits of the K-th largest key
  uint32_t remk   = TK_K;    // top-k slots remaining inside current prefix group
  uint32_t eqcnt  = 0;       // count of keys equal to the final threshold key

  for (int pass = 0; pass < 4; ++pass) {
    const int shift = 24 - pass * 8;
    // clear per-wave histograms
    for (int i = tid; i < TK_WAVES * 256; i += TK_T) whist[i] = 0;
    __syncthreads();

    uint32_t* myh = &whist[(tid >> 5) * 256];            // wave32-private copy
    const uint32_t pmask = (pass == 0) ? 0u : (0xFFFFFFFFu << (shift + 8));
    for (int i = tid; i < TK_N; i += TK_T) {
      const uint32_t key = keys[i];
      if ((key & pmask) == prefix)
        atomicAdd(&myh[(key >> shift) & 0xFFu], 1u);     // ds_add_u32
    }
    __syncthreads();

    // reduce the 32 wave histograms
    if (tid < 256) {
      uint32_t s = 0;
      #pragma unroll
      for (int w = 0; w < TK_WAVES; ++w) s += whist[w * 256 + tid];
      hist[tid] = s;
      suf[tid]  = s;
    }
    __syncthreads();

    // inclusive suffix sum from the top bin (descending selection)
    #pragma unroll
    for (int d = 1; d < 256; d <<= 1) {
      uint32_t v = 0;
      if (tid < 256) { v = suf[tid]; if (tid + d < 256) v += suf[tid + d]; }
      __syncthreads();
      if (tid < 256) suf[tid] = v;
      __syncthreads();
    }

    // unique crossing bin: suf[b] >= remk, suf[b+1] < remk
    if (tid < 256) {
      const uint32_t here  = suf[tid];
      const uint32_t above = (tid < 255) ? suf[tid + 1] : 0u;
      if (here >= remk && above < remk) {
        s_bin = (uint32_t)tid; s_above = above; s_bincnt = hist[tid];
      }
    }
    __syncthreads();

    prefix |= (s_bin << shift);
    remk   -= s_above;        // still to be taken from the chosen bin's subtree
    eqcnt   = s_bincnt;
    __syncthreads();
  }

  const uint32_t Kkey    = prefix;   // exact k-th largest key
  const uint32_t need_eq = remk;     // how many threshold-equal elems to keep

  // ---- Masked writeback: non-temporal, coalesced 16B per lane ----
  // Output is a write-once 805 MB stream; NT hint keeps it out of L2.
  #pragma unroll
  for (int c = 0; c < TK_V4; ++c) {
    const int v4 = tid + c * TK_T;
    v4f o;
    #pragma unroll
    for (int e = 0; e < 4; ++e) {
      const int gi = v4 * 4 + e;
      const uint32_t key = keys[gi];
      float val = 0.0f;
      if (key > Kkey) {
        val = key2f(key);
      } else if (key == Kkey) {
        if (need_eq == eqcnt) {
          val = key2f(key);                 // all equals are in the top-k
        } else {
          // tie on the threshold value: keep lowest indices (jax top_k order)
          uint32_t rank = 0;
          for (int j = 0; j < gi; ++j) rank += (keys[j] == Kkey) ? 1u : 0u;
          if (rank < need_eq) val = key2f(key);
        }
      }
      o[e] = val;
    }
    __builtin_nontemporal_store(o, (v4f*)(out + rowBase) + v4);
  }
}

extern "C" void kernel_launch(void* const* d_in, const int* in_sizes, int n_in,
                              void* d_out, int out_size, void* d_ws, size_t ws_size,
                              hipStream_t stream) {
  (void)n_in; (void)out_size; (void)d_ws; (void)ws_size;
  const float* x = (const float*)d_in[0];   // [rows, 24576] f32
  // d_in[1] is k (scalar 64) -- baked in at compile time
  float* out = (float*)d_out;
  const int rows = in_sizes[0] / TK_N;      // 8192
  hipLaunchKernelGGL(topk_mask_kernel, dim3(rows), dim3(TK_T), 0, stream, x, out);
}